// TopK_custom2_w_initial_31258771981051
// MI455X (gfx1250) — compile-verified
//
#include <hip/hip_runtime.h>
#include <math.h>

#define NN     4096
#define KTOP   15
#define KC     16
#define BSZ    128
#define EPSF   0.1f
#define IEPS   10.0f
#define SMALLF 1e-20f
#define MAXIT  200
#define MTOT   (NN*KC)   // 65536

// workspace layout (float offsets)
#define WS_FILLED 0
#define WS_INVC   1
#define WS_TV     16
#define WS_CI     (WS_TV + BSZ*16)
#define WS_XPART  (WS_CI + BSZ*16)           // 16 ksplits * 8 mtiles * 256
#define WS_NORMP  (WS_XPART + 16*8*256)      // 128 per-batch squared norms

typedef float v8f __attribute__((ext_vector_type(8)));
typedef float v2f __attribute__((ext_vector_type(2)));

// ---------------- K1: global min/max/has(-inf) -> filled, 1/Cmax ----------------
__global__ __launch_bounds__(1024) void k1_scan(const float* __restrict__ sc, float* __restrict__ ws) {
  __shared__ float smx[1024], smn[1024];
  __shared__ int   sng[1024];
  int t = threadIdx.x;
  float mx = -INFINITY, mn = INFINITY; int ng = 0;
  for (int i = t; i < BSZ*NN; i += 1024) {
    float v = sc[i];
    if (v == -INFINITY) ng = 1;
    else { mx = fmaxf(mx, v); mn = fminf(mn, v); }
  }
  smx[t]=mx; smn[t]=mn; sng[t]=ng; __syncthreads();
  for (int o=512;o>0;o>>=1){
    if (t<o){ smx[t]=fmaxf(smx[t],smx[t+o]); smn[t]=fminf(smn[t],smn[t+o]); sng[t]|=sng[t+o]; }
    __syncthreads();
  }
  if (t==0){
    float maxs=smx[0], mins=smn[0];
    float filled = mins - (maxs - mins);
    float lo = sng[0] ? filled : mins;   // min value actually present after -inf fill
    float hi = maxs;
    // C.max over data: (s-a)^2 convex in s, extremal at a in {0,15}, s in {lo,hi}
    float cm = fmaxf(fmaxf(lo*lo,(lo-15.f)*(lo-15.f)), fmaxf(hi*hi,(hi-15.f)*(hi-15.f)));
    ws[WS_FILLED]=filled; ws[WS_INVC]=1.0f/cm;
  }
}

// fast sigmoid term: rcp is single v_rcp_f32 (arg >= 1, well conditioned)
__device__ __forceinline__ float sigp(float tvc, float s, float invtau){
  float d = fabsf(tvc - s);
  return __builtin_amdgcn_rcpf(1.0f + __expf(d*invtau)) + SMALLF;
}

// ---------------- K2: per-row top-15 + Gamma0 column normalizers ----------------
__global__ __launch_bounds__(512) void k2_prep(const float* __restrict__ sc, const float* __restrict__ tau,
                                               float* __restrict__ ws) {
  __shared__ float s_sh[NN];
  __shared__ float tmp[NN];
  __shared__ float rv[512];
  __shared__ int   ri[512];
  __shared__ float tv_sh[KTOP];
  __shared__ float red[512];
  int b = blockIdx.x, t = threadIdx.x;
  float filled = ws[WS_FILLED];
  for (int j=0;j<NN/512;j++){ int i=t+j*512; float v=sc[b*NN+i]; if (v==-INFINITY) v=filled; s_sh[i]=v; tmp[i]=v; }
  __syncthreads();
  // 15 rounds of argmax extraction (values identical to lax.top_k order)
  for (int r=0;r<KTOP;r++){
    float bv=-INFINITY; int bi=0;
    for (int j=0;j<NN/512;j++){ int i=t+j*512; float v=tmp[i]; if (v>bv){bv=v;bi=i;} }
    rv[t]=bv; ri[t]=bi; __syncthreads();
    for (int o=256;o>0;o>>=1){
      if (t<o){ float v2=rv[t+o]; int i2=ri[t+o];
        if (v2>rv[t] || (v2==rv[t] && i2<ri[t])){ rv[t]=v2; ri[t]=i2; } }
      __syncthreads();
    }
    if (t==0){ tv_sh[r]=rv[0]; tmp[ri[0]]=-INFINITY; }
    __syncthreads();
  }
  float invtau = 1.0f/tau[0];
  float acc[KTOP];
  #pragma unroll
  for (int c=0;c<KTOP;c++) acc[c]=0.f;
  for (int j=0;j<NN/512;j++){
    float s=s_sh[t+j*512];
    #pragma unroll
    for (int c=0;c<KTOP;c++) acc[c] += sigp(tv_sh[c], s, invtau);
  }
  #pragma unroll
  for (int c=0;c<KTOP;c++){
    red[t]=acc[c]; __syncthreads();
    for (int o=256;o>0;o>>=1){ if (t<o) red[t]+=red[t+o]; __syncthreads(); }
    if (t==0) ws[WS_CI + b*16 + c] = 1.0f/(red[0]*(float)NN);
    __syncthreads();
  }
  if (t<KTOP) ws[WS_TV + b*16 + t] = tv_sh[t];
}

// bvec(b,m) = C(b,m) + eps*log(Gamma0(b,m)), recomputed on the fly
__device__ __forceinline__ float bvec_elem(const float* __restrict__ sc, int bb, int m,
                                           float filled, float invC, float invtau,
                                           const float* tvp, const float* cip){
  int i = m >> 4, c = m & 15;
  float s = sc[bb*NN + i];
  if (s == -INFINITY) s = filled;
  float g0;
  if (c < KTOP) {
    g0 = sigp(tvp[c], s, invtau) * cip[c];
  } else {
    float rs = 0.f;
    #pragma unroll
    for (int cc=0; cc<KTOP; cc++) rs += sigp(tvp[cc], s, invtau) * cip[cc];
    g0 = fminf(fmaxf(1.0f/(float)NN - rs, SMALLF), 1.0f - SMALLF);
  }
  float a = (float)(KTOP - c);
  float d = s - a;
  return d*d*invC + EPSF*__logf(g0);
}

// ---------------- K3: WMMA GEMM  X = bvec(128x65536) @ W2pad(65536x16) ----------------
// grid = 8 M-tiles * 16 K-splits, one wave each; deterministic fixed-order K accumulation.
__global__ __launch_bounds__(32) void k3_gemm(const float* __restrict__ sc, const float* __restrict__ w2,
                                              const float* __restrict__ tau, float* __restrict__ ws) {
  __shared__ float tvb[16*17];
  __shared__ float cib[16*17];
  int mt = blockIdx.x & 7, ks = blockIdx.x >> 3;
  int lane = threadIdx.x;
  for (int idx=lane; idx<256; idx+=32){
    int rr = idx>>4, cc = idx&15;
    tvb[rr*17+cc] = ws[WS_TV + (mt*16+rr)*16 + cc];
    cib[rr*17+cc] = ws[WS_CI + (mt*16+rr)*16 + cc];
  }
  __syncthreads();
  float filled = ws[WS_FILLED], invC = ws[WS_INVC];
  float invtau = 1.0f/tau[0];
  int row = lane & 15;            // A: M row, B: N col
  int kh  = (lane >> 4) * 2;      // K sub-offset per ISA layout
  int bb  = mt*16 + row;
  const float* tvp = &tvb[row*17];
  const float* cip = &cib[row*17];
  v8f acc = {0.f,0.f,0.f,0.f,0.f,0.f,0.f,0.f};
  int kbase = ks * (MTOT/16);
  for (int kk=0; kk<MTOT/16; kk+=4){
    int m0 = kbase + kk + kh;
    v2f av, bv;
    av.x = bvec_elem(sc, bb, m0,   filled, invC, invtau, tvp, cip);
    av.y = bvec_elem(sc, bb, m0+1, filled, invC, invtau, tvp, cip);
    bv.x = (row<5) ? w2[row*MTOT + m0]   : 0.f;   // W2pad[k][r] = weight2[r][k], rank padded to 16
    bv.y = (row<5) ? w2[row*MTOT + m0+1] : 0.f;
    acc = __builtin_amdgcn_wmma_f32_16x16x4_f32(false, av, false, bv, (short)0, acc, false, false);
  }
  int half = lane >> 4;
  #pragma unroll
  for (int v=0; v<8; v++){
    int M = v + 8*half;
    ws[WS_XPART + (ks*8+mt)*256 + M*16 + row] = acc[v];
  }
}

// ---------------- K4: 200 Sinkhorn iterations per batch row, all in LDS ----------------
__global__ __launch_bounds__(512) void k4_sinkhorn(const float* __restrict__ sc, const float* __restrict__ tau,
                                                   const float* __restrict__ w1, float* __restrict__ ws,
                                                   float* __restrict__ out) {
  __shared__ float s_sh[NN];
  __shared__ float f_sh[NN];
  __shared__ float g_sh[16];
  __shared__ float M_sh[16];
  __shared__ float pm[16*33];
  __shared__ float ps[16*33];
  __shared__ float tv_sh[16], ci_sh[16];
  __shared__ float x_sh[5];
  __shared__ float nred[512];
  int b = blockIdx.x, t = threadIdx.x;
  float filled = ws[WS_FILLED], invC = ws[WS_INVC];
  float invtau = 1.0f/tau[0];
  float k2 = invC * IEPS;
  const float elog_mu      = EPSF * __logf(1.0f/(float)NN);
  const float elog_nu_last = EPSF * __logf((float)(NN-KTOP)/(float)NN);
  for (int j=0;j<NN/512;j++){ int i=t+j*512; float v=sc[b*NN+i]; if (v==-INFINITY) v=filled; s_sh[i]=v; }
  if (t<16){ tv_sh[t]=ws[WS_TV+b*16+t]; ci_sh[t]=ws[WS_CI+b*16+t]; }
  if (t<5){ // fold GEMM K-split reduction (fixed order -> deterministic)
    float xr=0.f;
    for (int kss=0;kss<16;kss++) xr += ws[WS_XPART + (kss*8 + (b>>4))*256 + (b&15)*16 + t];
    x_sh[t]=xr;
  }
  __syncthreads();
  if (t<16){ // g init = sum_r X[b,r]*W1[n+c,r]; g_15 = 0. (f init is dead: scan overwrites f first.)
    float g=0.f;
    if (t<KTOP){ for (int r=0;r<5;r++) g += x_sh[r]*w1[(NN+t)*5+r]; }
    g_sh[t]=g;
  }
  __syncthreads();
  int cc = t & 15, ch = t >> 4;            // g-phase assignment: (anchor, 128-item chunk)
  float anc_c = (float)(KTOP - cc);
  for (int it=0; it<MAXIT; it++){
    // ---- f-update: f_i = -eps*lse_c((g_c - C_ic)/eps) + elog_mu
    float p[16], q[16];
    #pragma unroll
    for (int c=0;c<16;c++){ float a=(float)(KTOP-c); p[c]=g_sh[c]*IEPS - a*a*k2; q[c]=2.f*a*k2; }
    for (int j=0;j<NN/512;j++){
      int i=t+j*512; float s=s_sh[i]; float t0=s*s*k2;
      float arr[16]; float m=-INFINITY;
      #pragma unroll
      for (int c=0;c<16;c++){ float ag = fmaf(q[c], s, p[c]) - t0; arr[c]=ag; m=fmaxf(m,ag); }
      float sm=0.f;
      #pragma unroll
      for (int c=0;c<16;c++) sm += __expf(arr[c]-m);
      f_sh[i] = -EPSF*(m + __logf(sm)) + elog_mu;
    }
    __syncthreads();
    // ---- g-update: g_c = -eps*lse_i((f_i - C_ic)/eps) + elog_nu_c  (max pass + sum pass)
    float mx=-INFINITY;
    int ibase = ch*128;
    for (int jj=0;jj<128;jj++){
      int i=ibase+jj; float d=s_sh[i]-anc_c;
      float ag = f_sh[i]*IEPS - d*d*k2;
      mx = fmaxf(mx, ag);
    }
    pm[cc*33+ch]=mx; __syncthreads();
    if (t<16){ float m=-INFINITY; for (int h=0;h<32;h++) m=fmaxf(m,pm[t*33+h]); M_sh[t]=m; }
    __syncthreads();
    float Mc = M_sh[cc]; float sm=0.f;
    for (int jj=0;jj<128;jj++){
      int i=ibase+jj; float d=s_sh[i]-anc_c;
      float ag = f_sh[i]*IEPS - d*d*k2;
      sm += __expf(ag - Mc);
    }
    ps[cc*33+ch]=sm; __syncthreads();
    if (t<16){
      float s2=0.f; for (int h=0;h<32;h++) s2+=ps[t*33+h];
      float en = (t<KTOP)? elog_mu : elog_nu_last;
      g_sh[t] = -EPSF*(M_sh[t] + __logf(s2)) + en;
    }
    __syncthreads();
  }
  // ---- finalize: Gamma = exp((f+g-C)/eps); A = Gamma[:,:,:15]*n; accumulate ||Gamma-Gamma0||^2
  float lnorm = 0.f;
  for (int j=0;j<NN/512;j++){
    int i=t+j*512; float s=s_sh[i]; float fbase=f_sh[i]*IEPS;
    float g0[16]; float rs=0.f;
    #pragma unroll
    for (int c=0;c<KTOP;c++){ float gg = sigp(tv_sh[c], s, invtau)*ci_sh[c]; g0[c]=gg; rs+=gg; }
    g0[KTOP] = fminf(fmaxf(1.0f/(float)NN - rs, SMALLF), 1.0f-SMALLF);
    #pragma unroll
    for (int c=0;c<16;c++){
      float a=(float)(KTOP-c); float d=s-a;
      float gam = __expf(fbase + g_sh[c]*IEPS - d*d*k2);
      if (c<KTOP) out[((size_t)b*NN + i)*KTOP + c] = gam*(float)NN;
      float df = gam - g0[c]; lnorm += df*df;
    }
  }
  nred[t]=lnorm; __syncthreads();
  for (int o=256;o>0;o>>=1){ if (t<o) nred[t]+=nred[t+o]; __syncthreads(); }
  if (t==0) ws[WS_NORMP + b] = nred[0];
}

// ---------------- K5: deterministic final norm ----------------
__global__ void k5_finish(const float* __restrict__ ws, float* __restrict__ out){
  if (threadIdx.x==0 && blockIdx.x==0){
    float s=0.f;
    for (int b=0;b<BSZ;b++) s += ws[WS_NORMP+b];
    out[(size_t)BSZ*NN*KTOP] = sqrtf(s);
  }
}

extern "C" void kernel_launch(void* const* d_in, const int* in_sizes, int n_in,
                              void* d_out, int out_size, void* d_ws, size_t ws_size,
                              hipStream_t stream) {
  (void)in_sizes; (void)n_in; (void)out_size; (void)ws_size;
  const float* scores = (const float*)d_in[0];
  const float* tau    = (const float*)d_in[1];
  const float* w1     = (const float*)d_in[2];  // (N+K, 5)
  const float* w2     = (const float*)d_in[3];  // (5, N*16)
  float* out = (float*)d_out;
  float* ws  = (float*)d_ws;
  k1_scan    <<<1,   1024, 0, stream>>>(scores, ws);
  k2_prep    <<<BSZ, 512,  0, stream>>>(scores, tau, ws);
  k3_gemm    <<<128, 32,   0, stream>>>(scores, w2, tau, ws);
  k4_sinkhorn<<<BSZ, 512,  0, stream>>>(scores, tau, w1, ws, out);
  k5_finish  <<<1,   1,    0, stream>>>(ws, out);
}